// MultiHeadAttention_24979529793751
// MI455X (gfx1250) — compile-verified
//
#include <hip/hip_runtime.h>
#include <hip/hip_bf16.h>

// Shapes (fixed by the reference)
#define SLEN 256
#define BSZ 4
#define EMB 512
#define NHEAD 8
#define HD 64
// B = BSZ*NHEAD = 32 batched heads

typedef __attribute__((ext_vector_type(2))) float v2f;
typedef __attribute__((ext_vector_type(8))) float v8f;

// ---------------------------------------------------------------------------
// WMMA f32 16x16x4 helpers (CDNA5 VGPR layouts per ISA 7.12.2):
//   A 16x4 : lanes 0-15 -> M=lane, v0=K0,v1=K1 ; lanes 16-31 -> M=lane-16, v0=K2,v1=K3
//   B 4x16 : lanes 0-15 -> N=lane, v0=K0,v1=K1 ; lanes 16-31 -> N=lane-16, v0=K2,v1=K3
//   C 16x16: vgpr r: lanes0-15 M=r, lanes16-31 M=r+8; N=lane&15
// ---------------------------------------------------------------------------
__device__ __forceinline__ v8f wmma4(v2f a, v2f b, v8f c) {
  return __builtin_amdgcn_wmma_f32_16x16x4_f32(false, a, false, b, (short)0, c,
                                               false, false);
}

// "K-contiguous" fragment: element (idx16, k) at p[idx16*ld + k]; reads (k0,k0+1)
// for lanes<16 and (k0+2,k0+3) for lanes>=16 as one b64. Works for A row-major
// and for B when B[k][n] = M[n][k] (i.e. M row-major read along k).
__device__ __forceinline__ v2f ld_frag_k(const float* __restrict__ p, int ld) {
  int lane = threadIdx.x & 31;
  int l = lane & 15, half = lane >> 4;
  return *(const v2f*)(p + l * ld + half * 2);
}

// Row-major B fragment: B[k][n] at p[k*ld + n] (contiguous in n).
__device__ __forceinline__ v2f ld_frag_rm(const float* __restrict__ p, int ld) {
  int lane = threadIdx.x & 31;
  int l = lane & 15, half = lane >> 4;
  v2f r;
  r.x = p[(half * 2) * ld + l];
  r.y = p[(half * 2 + 1) * ld + l];
  return r;
}

__device__ __forceinline__ void st_tile(float* __restrict__ p, int ld, v8f c) {
  int lane = threadIdx.x & 31;
  int l = lane & 15, half = lane >> 4;
#pragma unroll
  for (int r = 0; r < 8; ++r) p[(half * 8 + r) * ld + l] = c[r];
}

// ---------------------------------------------------------------------------
// Kernel 1/5: Out[1024x512] = X[1024x512] @ W^T  (W row-major 512x512)
// grid 256 blocks x 256 thr (8 waves): 2048 tiles of 16x16
// ---------------------------------------------------------------------------
__global__ __launch_bounds__(256) void gemm_xwT(const float* __restrict__ X,
                                                const float* __restrict__ W,
                                                float* __restrict__ Out) {
  int tile = blockIdx.x * 8 + (threadIdx.x >> 5);
  int mt = tile >> 5, nt = tile & 31;  // 64 m-tiles, 32 n-tiles
  const float* Xp = X + mt * 16 * 512;
  const float* Wp = W + nt * 16 * 512;
  v8f acc = {};
#pragma unroll 8
  for (int kk = 0; kk < 512; kk += 4) {
    v2f a = ld_frag_k(Xp + kk, 512);
    v2f b = ld_frag_k(Wp + kk, 512);  // B[k][n] = W[n][k]
    acc = wmma4(a, b, acc);
  }
  st_tile(Out + mt * 16 * 512 + nt * 16, 512, acc);
}

// ---------------------------------------------------------------------------
// Kernel 2: Akn[g][i][h][e] = sum_d Q[(i*4+g)][h*64+d] * Wkn[h*64+d][e]
// grid (64, 32): blockIdx.y = g*8+h ; 512 tiles per head
// ---------------------------------------------------------------------------
__global__ __launch_bounds__(256) void akn_kernel(const float* __restrict__ Q,
                                                  const float* __restrict__ Wkn,
                                                  float* __restrict__ Akn) {
  int tile = blockIdx.x * 8 + (threadIdx.x >> 5);
  int mt = tile >> 5, nt = tile & 31;  // 16 i-tiles, 32 e-tiles
  int g = blockIdx.y >> 3, h = blockIdx.y & 7;
  const float* Ap = Q + (size_t)(mt * 16 * 4 + g) * 512 + h * 64;  // i-stride 2048
  const float* Bp = Wkn + (size_t)(h * 64) * 512 + nt * 16;        // row-major B[d][e]
  v8f acc = {};
#pragma unroll 4
  for (int kk = 0; kk < 64; kk += 4) {
    v2f a = ld_frag_k(Ap + kk, 2048);
    v2f b = ld_frag_rm(Bp + (size_t)kk * 512, 512);
    acc = wmma4(a, b, acc);
  }
  st_tile(Akn + ((size_t)(g * 256 + mt * 16) * 8 + h) * 512 + nt * 16, 8 * 512, acc);
}

// ---------------------------------------------------------------------------
// Kernel 3: logits0[b][i][j] = sum_d Q[b,i,d]*K[b,j,d]   (batched QK^T)
// grid (32, 32): blockIdx.y = b ; 256 tiles per batch
// ---------------------------------------------------------------------------
__global__ __launch_bounds__(256) void qk_kernel(const float* __restrict__ Q,
                                                 const float* __restrict__ K,
                                                 float* __restrict__ L0) {
  int tile = blockIdx.x * 8 + (threadIdx.x >> 5);
  int mt = tile >> 4, nt = tile & 15;
  int b = blockIdx.y, g = b >> 3, h = b & 7;
  const float* Ap = Q + (size_t)(mt * 16 * 4 + g) * 512 + h * 64;
  const float* Bp = K + (size_t)(nt * 16 * 4 + g) * 512 + h * 64;  // B[d][j]=K[j][d], j-stride 2048
  v8f acc = {};
#pragma unroll 4
  for (int kk = 0; kk < 64; kk += 4) {
    v2f a = ld_frag_k(Ap + kk, 2048);
    v2f b2 = ld_frag_k(Bp + kk, 2048);
    acc = wmma4(a, b2, acc);
  }
  st_tile(L0 + (size_t)b * 65536 + mt * 16 * 256 + nt * 16, 256, acc);
}

// ---------------------------------------------------------------------------
// Kernel 4 (HOT, streams `know` once = 512MB): per block (g,i):
//   knlog[j][h] = sum_e know[g,i,j,e] * Akn[g,i,h,e]   via WMMA (N=16, 8 live)
//   then logits=(knlog+L0)*0.125 - (1-adj)*1e30; softmax over j;
//   wS = w*s_mask, wO = w*o_mask
// ---------------------------------------------------------------------------
#define LSTR 272  // padded logits row stride
__global__ __launch_bounds__(256) void know_attn_kernel(
    const float* __restrict__ know, const float* __restrict__ Akn,
    const float* __restrict__ L0, const float* __restrict__ adj,
    const float* __restrict__ smask, const float* __restrict__ omask,
    float* __restrict__ wS, float* __restrict__ wO) {
  __shared__ __align__(16) float bfrag[128 * 64];  // pre-swizzled B fragments, 32KB
  __shared__ float lgt[8 * LSTR];                  // kn-logits [h][j]
  int i = blockIdx.x, g = blockIdx.y;
  int t = threadIdx.x;
  int wv = t >> 5, lane = t & 31, l = lane & 15, half = lane >> 4;

  // --- C1: build fragment-ordered B (Akn cols 0..7 = heads, 8..15 = zero pad)
  const float* akp = Akn + (size_t)(g * 256 + i) * 4096;  // [h][e]
  for (int idx = t; idx < 4096; idx += 256) {
    int ks = idx >> 5, ln = idx & 31;
    int nl = ln & 15, hf = ln >> 4;
    int e = ks * 4 + hf * 2;
    float v0 = 0.f, v1 = 0.f;
    if (nl < 8) {
      v0 = akp[nl * 512 + e];
      v1 = akp[nl * 512 + e + 1];
    }
    bfrag[ks * 64 + ln * 2] = v0;
    bfrag[ks * 64 + ln * 2 + 1] = v1;
  }
  __syncthreads();

  // --- C2: WMMA over 256x512 know panel; wave owns j-tiles {wv, wv+8}
  const float* krow = know + (size_t)(g * 256 + i) * 256 * 512;
#pragma unroll 1
  for (int pass = 0; pass < 2; ++pass) {
    int mt = wv + pass * 8;
    const float* Ap = krow + (size_t)mt * 16 * 512;
    v8f acc = {};
#pragma unroll 8
    for (int ks = 0; ks < 128; ++ks) {
      v2f a = ld_frag_k(Ap + ks * 4, 512);
      v2f b = *(const v2f*)(&bfrag[ks * 64 + lane * 2]);
      acc = wmma4(a, b, acc);
    }
    if (l < 8) {
#pragma unroll
      for (int r = 0; r < 8; ++r) lgt[l * LSTR + mt * 16 + half * 8 + r] = acc[r];
    }
  }
  __syncthreads();

  // --- C3: per-head softmax (wave wv <-> head h), masks, store weights
  int h = wv;
  int b = g * 8 + h;
  size_t base = (size_t)b * 65536 + (size_t)i * 256;
  float lr[8];
#pragma unroll
  for (int c = 0; c < 8; ++c) {
    int j = c * 32 + lane;
    float kn = lgt[h * LSTR + j];
    float l0 = L0[base + j];
    float av = adj[base + j];
    lr[c] = (kn + l0) * 0.125f - (1.0f - av) * 1e30f;
  }
  float m = lr[0];
#pragma unroll
  for (int c = 1; c < 8; ++c) m = fmaxf(m, lr[c]);
  for (int off = 16; off >= 1; off >>= 1) m = fmaxf(m, __shfl_xor(m, off, 32));
  float s = 0.f;
#pragma unroll
  for (int c = 0; c < 8; ++c) {
    lr[c] = __expf(lr[c] - m);
    s += lr[c];
  }
  for (int off = 16; off >= 1; off >>= 1) s += __shfl_xor(s, off, 32);
  float inv = 1.0f / s;
#pragma unroll
  for (int c = 0; c < 8; ++c) {
    int j = c * 32 + lane;
    float w = lr[c] * inv;
    float sm = smask[base + j];
    float om = omask[base + j];
    wS[base + j] = w * sm;
    wO[base + j] = w * om;
  }
}

// ---------------------------------------------------------------------------
// Kernel 5: attn[(i*4+g)*512 + h*64+d] = sum_j wS*Vs + wO*Vo   (batched)
// grid (8, 32): blockIdx.y = b ; 64 tiles per batch (16 i-tiles x 4 d-tiles)
// ---------------------------------------------------------------------------
__global__ __launch_bounds__(256) void attn_kernel(const float* __restrict__ wS,
                                                   const float* __restrict__ wO,
                                                   const float* __restrict__ Vs,
                                                   const float* __restrict__ Vo,
                                                   float* __restrict__ attn) {
  int tile = blockIdx.x * 8 + (threadIdx.x >> 5);
  int mt = tile >> 2, nt = tile & 3;
  int b = blockIdx.y, g = b >> 3, h = b & 7;
  const float* As = wS + (size_t)b * 65536 + mt * 16 * 256;
  const float* Ao = wO + (size_t)b * 65536 + mt * 16 * 256;
  const float* Bs = Vs + (size_t)g * 512 + h * 64 + nt * 16;  // B[j][d], j-stride 2048
  const float* Bo = Vo + (size_t)g * 512 + h * 64 + nt * 16;
  v8f acc = {};
#pragma unroll 2
  for (int kk = 0; kk < 256; kk += 4) {
    v2f a1 = ld_frag_k(As + kk, 256);
    v2f b1 = ld_frag_rm(Bs + (size_t)kk * 2048, 2048);
    acc = wmma4(a1, b1, acc);
    v2f a2 = ld_frag_k(Ao + kk, 256);
    v2f b2 = ld_frag_rm(Bo + (size_t)kk * 2048, 2048);
    acc = wmma4(a2, b2, acc);
  }
  st_tile(attn + (size_t)(mt * 16 * 4 + g) * 512 + h * 64 + nt * 16, 2048, acc);
}

// ---------------------------------------------------------------------------
extern "C" void kernel_launch(void* const* d_in, const int* in_sizes, int n_in,
                              void* d_out, int out_size, void* d_ws, size_t ws_size,
                              hipStream_t stream) {
  const float* x = (const float*)d_in[0];
  const float* adj = (const float*)d_in[1];
  const float* smask = (const float*)d_in[2];
  const float* omask = (const float*)d_in[3];
  const float* know = (const float*)d_in[4];
  const float* Wq = (const float*)d_in[5];
  const float* Wk = (const float*)d_in[6];
  const float* Wvs = (const float*)d_in[7];
  const float* Wvo = (const float*)d_in[8];
  const float* Wo = (const float*)d_in[9];
  const float* Wkn = (const float*)d_in[10];
  float* out = (float*)d_out;

  float* ws = (float*)d_ws;
  float* Q = ws;                  // 1024x512
  float* K = ws + 524288;         // 1024x512
  float* Vs = ws + 1048576;       // 1024x512
  float* Vo = ws + 1572864;       // 1024x512
  float* Akn = ws + 2097152;      // [g][i][h][e] = 4x256x8x512
  float* L0 = ws + 6291456;       // [b][i][j] = 32x256x256
  float* wS = ws + 8388608;       // 32x256x256
  float* wO = ws + 10485760;      // 32x256x256
  float* attn = ws + 12582912;    // 1024x512   (end: 13107200 floats = 52.4MB)

  // projections
  gemm_xwT<<<256, 256, 0, stream>>>(x, Wq, Q);
  gemm_xwT<<<256, 256, 0, stream>>>(x, Wk, K);
  gemm_xwT<<<256, 256, 0, stream>>>(x, Wvs, Vs);
  gemm_xwT<<<256, 256, 0, stream>>>(x, Wvo, Vo);
  // Akn = q @ Wkn (per-head), logits0 = Q K^T
  akn_kernel<<<dim3(64, 32), 256, 0, stream>>>(Q, Wkn, Akn);
  qk_kernel<<<dim3(32, 32), 256, 0, stream>>>(Q, K, L0);
  // fused know contraction + mask + softmax
  know_attn_kernel<<<dim3(256, 4), 256, 0, stream>>>(know, Akn, L0, adj, smask,
                                                     omask, wS, wO);
  // attention-weighted values, then output projection
  attn_kernel<<<dim3(8, 32), 256, 0, stream>>>(wS, wO, Vs, Vo, attn);
  gemm_xwT<<<256, 256, 0, stream>>>(attn, Wo, out);
}